// RGCN_17806934409351
// MI455X (gfx1250) — compile-verified
//
#include <hip/hip_runtime.h>

// ---------------- problem constants (from reference) ----------------
#define HD   256           // hidden dim H
#define NB   128           // number of block-diagonal bases B
#define RT   16            // number of relations R
#define WTAB (RT * NB * 4) // 16*128*(2*2) = 8192 floats = 32 KB relation table

typedef float v2f __attribute__((ext_vector_type(2)));
typedef float v8f __attribute__((ext_vector_type(8)));

// ====================================================================
// Self-loop GEMM with bias:  out[n, j] = act(A[n, :]) @ W[:, j] + bias[j]
// One wave32 computes a 16x64 output strip: the A fragment (16x4 f32)
// is loaded once per K-step and reused by 4 V_WMMA_F32_16X16X4_F32
// against 4 B fragments (4x16 each).  RELU is a compile-time flag so
// layer 1 carries zero activation overhead in the K-loop.
// ====================================================================
template <bool RELU>
__global__ __launch_bounds__(256)
void rgcn_selfloop_gemm(const float* __restrict__ A,
                        const float* __restrict__ W,
                        const float* __restrict__ bias,
                        float* __restrict__ out,
                        int n_rows)
{
    const int lane = threadIdx.x & 31;
    const int wid  = blockIdx.x * (blockDim.x >> 5) + (threadIdx.x >> 5);

    const int strips_n = HD / 64;            // 4 strips of 64 cols
    const int tile_m   = wid / strips_n;
    const int strip_n  = wid % strips_n;
    if (tile_m * 16 >= n_rows) return;       // wave-uniform guard

    const int half = lane >> 4;              // 0: K pair {0,1}, 1: K pair {2,3}
    const int sub  = lane & 15;
    const int arow_idx = tile_m * 16 + sub;          // A row (M = sub)
    const int col0     = strip_n * 64 + sub;         // first of 4 output cols

    const float* arow = A + (size_t)arow_idx * HD;

    v8f c0 = {}, c1 = {}, c2 = {}, c3 = {};
    for (int k = 0; k < HD; k += 4) {
        // A fragment (16x4 f32): VGPR0 = K = 2*half, VGPR1 = K = 2*half+1
        float a0 = arow[k + 2 * half + 0];
        float a1 = arow[k + 2 * half + 1];
        if (RELU) { a0 = fmaxf(a0, 0.0f); a1 = fmaxf(a1, 0.0f); }
        v2f a; a.x = a0; a.y = a1;

        // 4 B fragments (4x16 f32 each), cols col0 + {0,16,32,48}
        const float* wrow0 = W + (size_t)(k + 2 * half + 0) * HD + col0;
        const float* wrow1 = W + (size_t)(k + 2 * half + 1) * HD + col0;
        v2f b0, b1, b2, b3;
        b0.x = wrow0[0];  b0.y = wrow1[0];
        b1.x = wrow0[16]; b1.y = wrow1[16];
        b2.x = wrow0[32]; b2.y = wrow1[32];
        b3.x = wrow0[48]; b3.y = wrow1[48];

        // D = A*B + C   (neg_a, A, neg_b, B, c_mod, C, reuse_a, reuse_b)
        c0 = __builtin_amdgcn_wmma_f32_16x16x4_f32(false, a, false, b0, (short)0, c0, false, false);
        c1 = __builtin_amdgcn_wmma_f32_16x16x4_f32(false, a, false, b1, (short)0, c1, false, false);
        c2 = __builtin_amdgcn_wmma_f32_16x16x4_f32(false, a, false, b2, (short)0, c2, false, false);
        c3 = __builtin_amdgcn_wmma_f32_16x16x4_f32(false, a, false, b3, (short)0, c3, false, false);
    }

    // C/D layout: VGPR j -> row (tile_m*16 + half*8 + j), col = col0 + t*16
    float* obase = out + (size_t)(tile_m * 16 + half * 8) * HD + col0;
    #pragma unroll
    for (int t = 0; t < 4; ++t) {
        const v8f& c = (t == 0) ? c0 : (t == 1) ? c1 : (t == 2) ? c2 : c3;
        const float bcol = bias[col0 + t * 16];
        #pragma unroll
        for (int j = 0; j < 8; ++j)
            obase[(size_t)j * HD + t * 16] = c[j] + bcol;
    }
}

// ====================================================================
// Edge message + scatter-add:
//   agg[dst[e], :] += norm[e] * blockdiag(w[etype[e]]) @ act(x[src[e], :])
// Relation table (32 KB) staged in LDS.  One wave per edge (grid-stride);
// lane L owns bases 4L..4L+3 => x elements 8L..8L+7 (two float4 loads),
// 8 f32 outputs scattered with global_atomic_add_f32.
// ====================================================================
template <bool RELU>
__global__ __launch_bounds__(256)
void rgcn_edge_msg(const float* __restrict__ xin,
                   const int*   __restrict__ src,
                   const int*   __restrict__ dst,
                   const int*   __restrict__ etype,
                   const float* __restrict__ norm,
                   const float* __restrict__ w,      // (R, B, 2, 2) row-major
                   float* agg,
                   int n_edges)
{
    __shared__ __align__(16) float wlds[WTAB];
    for (int i = threadIdx.x; i < WTAB; i += blockDim.x) wlds[i] = w[i];
    __syncthreads();

    const int lane   = threadIdx.x & 31;
    const int wid    = blockIdx.x * (blockDim.x >> 5) + (threadIdx.x >> 5);
    const int nwaves = (gridDim.x * blockDim.x) >> 5;

    for (int e = wid; e < n_edges; e += nwaves) {
        const int   s   = src[e];
        const int   d   = dst[e];
        const int   r   = etype[e];
        const float nrm = norm[e];

        // source row: lane covers 8 consecutive floats (bases 4L..4L+3)
        const float4* xrow = (const float4*)(xin + (size_t)s * HD) + lane * 2;
        float4 xa = xrow[0];
        float4 xb = xrow[1];
        if (RELU) {
            xa.x = fmaxf(xa.x, 0.0f); xa.y = fmaxf(xa.y, 0.0f);
            xa.z = fmaxf(xa.z, 0.0f); xa.w = fmaxf(xa.w, 0.0f);
            xb.x = fmaxf(xb.x, 0.0f); xb.y = fmaxf(xb.y, 0.0f);
            xb.z = fmaxf(xb.z, 0.0f); xb.w = fmaxf(xb.w, 0.0f);
        }
        const float xs[8] = { xa.x, xa.y, xa.z, xa.w, xb.x, xb.y, xb.z, xb.w };

        // relation blocks from LDS: base b is 4 contiguous floats [w00,w01,w10,w11]
        const float4* wrel = (const float4*)(wlds + (size_t)r * (NB * 4)) + lane * 4;
        float* arow = agg + (size_t)d * HD + lane * 8;

        #pragma unroll
        for (int j = 0; j < 4; ++j) {
            const float4 wq = wrel[j];
            const float x0 = xs[2 * j + 0];
            const float x1 = xs[2 * j + 1];
            const float o0 = (x0 * wq.x + x1 * wq.z) * nrm;   // out_0 = x0*w00 + x1*w10
            const float o1 = (x0 * wq.y + x1 * wq.w) * nrm;   // out_1 = x0*w01 + x1*w11
            (void)__hip_atomic_fetch_add(&arow[2 * j + 0], o0,
                                         __ATOMIC_RELAXED, __HIP_MEMORY_SCOPE_AGENT);
            (void)__hip_atomic_fetch_add(&arow[2 * j + 1], o1,
                                         __ATOMIC_RELAXED, __HIP_MEMORY_SCOPE_AGENT);
        }
    }
}

// ====================================================================
// Launch: layer1 -> h (workspace), layer2 -> d_out.  GEMM writes the
// aggregation buffer first (bias included), edge kernel accumulates
// on top; ReLU fused into all layer-2 input loads.
// ====================================================================
extern "C" void kernel_launch(void* const* d_in, const int* in_sizes, int n_in,
                              void* d_out, int out_size, void* d_ws, size_t ws_size,
                              hipStream_t stream)
{
    const float* x     = (const float*)d_in[0];
    const int*   src   = (const int*)  d_in[1];
    const int*   dst   = (const int*)  d_in[2];
    const int*   etype = (const int*)  d_in[3];
    const float* norm  = (const float*)d_in[4];
    const float* w1    = (const float*)d_in[5];
    const float* lw1   = (const float*)d_in[6];
    const float* b1    = (const float*)d_in[7];
    const float* w2    = (const float*)d_in[8];
    const float* lw2   = (const float*)d_in[9];
    const float* b2    = (const float*)d_in[10];
    float*       out   = (float*)d_out;

    const int n_nodes = in_sizes[0] / HD;   // 20000
    const int n_edges = in_sizes[1];        // 320000

    float* h = (float*)d_ws;                // n_nodes * HD floats (20.48 MB)

    const int waves_per_block = 256 / 32;
    const int strips = ((n_nodes + 15) / 16) * (HD / 64);   // 16-row x 64-col strips
    const int gemm_blocks = (strips + waves_per_block - 1) / waves_per_block;
    const int edge_blocks = 1024;           // 8192 waves grid-striding 320k edges

    // ----- layer 1:  h = b1 + x@lw1  (GEMM) ; h += scatter(msg(x,w1)) -----
    rgcn_selfloop_gemm<false><<<gemm_blocks, 256, 0, stream>>>(x, lw1, b1, h, n_nodes);
    rgcn_edge_msg<false><<<edge_blocks, 256, 0, stream>>>(x, src, dst, etype, norm, w1,
                                                          h, n_edges);

    // ----- layer 2 on relu(h), ReLU fused into loads -----
    rgcn_selfloop_gemm<true><<<gemm_blocks, 256, 0, stream>>>(h, lw2, b2, out, n_nodes);
    rgcn_edge_msg<true><<<edge_blocks, 256, 0, stream>>>(h, src, dst, etype, norm, w2,
                                                         out, n_edges);
}